// FeatureTransformerSlice_16441134809367
// MI455X (gfx1250) — compile-verified
//
#include <hip/hip_runtime.h>
#include <stdint.h>

// FeatureTransformerSlice: out[b,:] = sum_f val[b,f] * W[idx[b,f],:] + bias
// B=16384, F=32, O=512, W: [40960, 512] f32.
// Bandwidth-bound gather-reduce; W (80MB) is L2-resident (192MB L2).
// CDNA5 path: TDM gather-mode TENSOR_LOAD_TO_LDS (16 rows by 16-bit row index
// per descriptor, tracked by TENSORcnt) + s_wait_tensorcnt; wave32 layout.

#define FT_F       32
#define FT_OUT     512
#define FT_THREADS 256

typedef __attribute__((ext_vector_type(4))) unsigned v4u;
typedef __attribute__((ext_vector_type(8))) int      v8i;
typedef __attribute__((ext_vector_type(4))) int      v4i;

#if defined(__AMDGCN__) && __has_builtin(__builtin_amdgcn_tensor_load_to_lds)
#define FT_TDM 1
#else
#define FT_TDM 0
#endif

__global__ __launch_bounds__(FT_THREADS) void
FeatureTransformerSlice_16441134809367_kernel(const int* __restrict__ fidx,
                                              const float* __restrict__ fval,
                                              const float* __restrict__ weight,
                                              const float* __restrict__ bias,
                                              float* __restrict__ out,
                                              int num_inputs) {
  // 32 gathered rows x 512 f32 = 64 KB LDS tile + 128 B staged values.
  __shared__ float tile[FT_F * FT_OUT];
  __shared__ float vals[FT_F];

  const int b   = blockIdx.x;
  const int tid = threadIdx.x;

#if FT_TDM
  // Uniform (SGPR) wave id -> scalar s_cbranch, no exec divergence, no
  // lane-state preserved across the descriptor-building block.
  const int wv = __builtin_amdgcn_readfirstlane((int)tid) >> 5;

  if (wv < 2) {
    // --- CDNA5 Tensor Data Mover, gather mode -----------------------------
    // Wave 0 gathers rows 0..15 -> tile[0..]; wave 1 rows 16..31 -> +32KB.
    // Pure SALU descriptor build: one s_load_b512 of this wave's 16 indices,
    // clamp + pack as 16-bit pairs into 8 SGPR dwords (D# groups 2+3).
    const int  base_f = wv * 16;
    const int* ip     = fidx + (size_t)b * FT_F + base_f;

    int raw[16];
#pragma unroll
    for (int i = 0; i < 16; ++i) raw[i] = ip[i];  // uniform addr -> s_load

    unsigned p[8];
#pragma unroll
    for (int i = 0; i < 8; ++i) {
      const unsigned lo = (unsigned)(raw[2 * i] < 0 ? 0 : raw[2 * i]) & 0xFFFFu;
      const unsigned hi = (unsigned)(raw[2 * i + 1] < 0 ? 0 : raw[2 * i + 1]) & 0xFFFFu;
      p[i] = lo | (hi << 16);
    }

    // LDS byte offset: low 32 bits of flat shared address == LDS offset.
    const unsigned lds_addr =
        (unsigned)(uintptr_t)(&tile[0]) + (unsigned)(base_f * FT_OUT * 4);
    const uint64_t ga = (uint64_t)(uintptr_t)weight;

    // D# group 0: count=1, gather_mode=1 (bit31), 16-bit indices (bit30=0),
    // lds_addr, global_addr[56:0], type=2 ("image") in bits [127:126].
    v4u g0;
    g0.x = 0x80000001u;
    g0.y = lds_addr;
    g0.z = (unsigned)(ga & 0xFFFFFFFFu);
    g0.w = (unsigned)((ga >> 32) & 0x01FFFFFFu) | (2u << 30);

    // D# group 1: data_size=2 (4B), tensor_dim0=512, tensor_dim1=num_inputs,
    // tile_dim0=512 (full row), tile_dim1=16 (#valid indices),
    // tensor_dim0_stride=512, workgroup_mask=0 (not in a cluster).
    const unsigned ni = (unsigned)num_inputs;
    v8i g1;
    g1[0] = (int)(2u << 16);
    g1[1] = (int)((unsigned)FT_OUT << 16);                             // dim0 lo
    g1[2] = (int)((ni & 0xFFFFu) << 16);                               // dim1 lo
    g1[3] = (int)(((ni >> 16) & 0xFFFFu) | ((unsigned)FT_OUT << 16));  // dim1 hi | tile_dim0
    g1[4] = 16;                                                        // tile_dim1
    g1[5] = FT_OUT;                                                    // dim0_stride
    g1[6] = 0;
    g1[7] = 0;

    v4i g2 = {(int)p[0], (int)p[1], (int)p[2], (int)p[3]};
    v4i g3 = {(int)p[4], (int)p[5], (int)p[6], (int)p[7]};

#if __clang_major__ >= 23
    v8i zero8 = {0, 0, 0, 0, 0, 0, 0, 0};
    __builtin_amdgcn_tensor_load_to_lds(g0, g1, g2, g3, zero8, 0);
#else
    __builtin_amdgcn_tensor_load_to_lds(g0, g1, g2, g3, 0);
#endif
  } else if (wv == 2) {
    // Wave 2: stage masked feature values into LDS with per-lane loads
    // (values never occupy SGPRs -> no scalar-register pressure).
    const int   lane = tid & 31;
    const int   raw  = fidx[(size_t)b * FT_F + lane];
    const float v    = fval[(size_t)b * FT_F + lane];
    vals[lane] = raw < 0 ? 0.0f : v;
  }
  // Issuing waves drain TENSORcnt (no-op for the others), then all sync.
  __builtin_amdgcn_s_wait_tensorcnt(0);
  __syncthreads();
#else
  // --- Fallback: cooperative VMEM gather into LDS -------------------------
  if (tid < FT_F) {
    const int   raw = fidx[(size_t)b * FT_F + tid];
    const float v   = fval[(size_t)b * FT_F + tid];
    vals[tid] = raw < 0 ? 0.0f : v;
  }
  for (int f = 0; f < FT_F; ++f) {
    const int raw = fidx[(size_t)b * FT_F + f];
    const int cl  = raw < 0 ? 0 : raw;
    for (int o = tid; o < FT_OUT; o += FT_THREADS)
      tile[f * FT_OUT + o] = weight[(size_t)cl * FT_OUT + o];
  }
  __syncthreads();
#endif

  // Each thread owns a float2 output pair: 32 conflict-free ds_load_b64
  // (64 banks, 32 lanes x 8B) + broadcast value reads + FMAs.
  const int o0   = tid * 2;
  float     acc0 = bias[o0];
  float     acc1 = bias[o0 + 1];
#pragma unroll
  for (int f = 0; f < FT_F; ++f) {
    const float  vv = vals[f];
    const float2 w2 = *reinterpret_cast<const float2*>(&tile[f * FT_OUT + o0]);
    acc0 = fmaf(w2.x, vv, acc0);
    acc1 = fmaf(w2.y, vv, acc1);
  }
  float2 r;
  r.x = acc0;
  r.y = acc1;
  *reinterpret_cast<float2*>(&out[(size_t)b * FT_OUT + o0]) = r;
}

extern "C" void kernel_launch(void* const* d_in, const int* in_sizes, int n_in,
                              void* d_out, int out_size, void* d_ws, size_t ws_size,
                              hipStream_t stream) {
  const int*   fidx   = (const int*)d_in[0];    // feature_indices [B, F]
  const float* fval   = (const float*)d_in[1];  // feature_values  [B, F]
  const float* weight = (const float*)d_in[2];  // [NUM_INPUTS, 512]
  const float* bias   = (const float*)d_in[3];  // [512]
  float*       out    = (float*)d_out;          // [B, 512]

  const int B          = in_sizes[1] / FT_F;
  const int num_inputs = in_sizes[2] / FT_OUT;

  dim3 grid((unsigned)B), block(FT_THREADS);
  hipLaunchKernelGGL(FeatureTransformerSlice_16441134809367_kernel, grid, block,
                     0, stream, fidx, fval, weight, bias, out, num_inputs);
}